// TransformerDynamics_2_83279415869557
// MI455X (gfx1250) — compile-verified
//
#include <hip/hip_runtime.h>
#include <math.h>

typedef float v2f __attribute__((ext_vector_type(2)));
typedef float v4f __attribute__((ext_vector_type(4)));
typedef float v8f __attribute__((ext_vector_type(8)));

#define BS  16
#define NN  512
#define NF  7
#define HID 64
#define EPSF 1e-8f

// ---------------------------------------------------------------------------
// Kernel A: remove_mean_with_mask -> write xc into rec[node][0..2]
// rec layout per node (8 floats, 32B): {xc0, xc1, xc2, he0, he1, pad, pad, pad}
// ---------------------------------------------------------------------------
__global__ void center_kernel(const float* __restrict__ x,
                              const float* __restrict__ mask,
                              float* __restrict__ rec) {
    const int b = blockIdx.x;          // 0..15
    const int t = threadIdx.x;         // 0..511 (16 waves)
    const float* xb = x + (size_t)b * NN * 3;
    float x0 = xb[t * 3 + 0];
    float x1 = xb[t * 3 + 1];
    float x2 = xb[t * 3 + 2];
    float mv = mask[b * NN + t];

    float s0 = x0 * mv, s1 = x1 * mv, s2 = x2 * mv, s3 = mv;
    // wave32 butterfly reduce
    #pragma unroll
    for (int off = 16; off >= 1; off >>= 1) {
        s0 += __shfl_xor(s0, off, 32);
        s1 += __shfl_xor(s1, off, 32);
        s2 += __shfl_xor(s2, off, 32);
        s3 += __shfl_xor(s3, off, 32);
    }
    __shared__ float red[16][4];
    __shared__ float meanv[3];
    const int wave = t >> 5, lane = t & 31;
    if (lane == 0) { red[wave][0] = s0; red[wave][1] = s1; red[wave][2] = s2; red[wave][3] = s3; }
    __syncthreads();
    if (t == 0) {
        float a0 = 0.f, a1 = 0.f, a2 = 0.f, a3 = 0.f;
        #pragma unroll
        for (int w = 0; w < 16; ++w) { a0 += red[w][0]; a1 += red[w][1]; a2 += red[w][2]; a3 += red[w][3]; }
        float inv = 1.0f / a3;          // once per block: precise div is fine
        meanv[0] = a0 * inv; meanv[1] = a1 * inv; meanv[2] = a2 * inv;
    }
    __syncthreads();
    float* r = rec + (size_t)(b * NN + t) * 8;
    r[0] = (x0 - meanv[0]) * mv;
    r[1] = (x1 - meanv[1]) * mv;
    r[2] = (x2 - meanv[2]) * mv;
}

// ---------------------------------------------------------------------------
// Kernel B: he = silu(h @ W1 + b1) @ W2 + b2  -> rec[node][3..4]
// Layer 1 via V_WMMA_F32_16X16X4_F32: per wave, one 16-row tile,
// 4 N-tiles x 2 K-steps (K=7 zero-padded to 8). Layer 2 via LDS + VALU
// (only 2 output columns). 256 threads = 8 waves per block.
// ---------------------------------------------------------------------------
__global__ void mlp_kernel(const float* __restrict__ h,
                           const float* __restrict__ W1,
                           const float* __restrict__ b1v,
                           const float* __restrict__ W2,
                           const float* __restrict__ b2v,
                           float* __restrict__ rec) {
    __shared__ float lds[8][16 * 68];   // 68-float row stride avoids bank conflicts

    const int lane = threadIdx.x & 31;
    const int wave = threadIdx.x >> 5;
    const int tile = blockIdx.x * 8 + wave;
    const int row0 = tile * 16;
    const int half = lane >> 4;         // 0: lanes 0-15, 1: lanes 16-31
    const int lm   = lane & 15;

    // A fragments (ISA: 16x4 f32 A -> lanes 0-15 hold K=0,1; lanes 16-31 K=2,3)
    const float* hrow = h + (size_t)(row0 + lm) * NF;
    const int k0 = half * 2;            // 0 or 2
    const int k1 = 4 + half * 2;        // 4 or 6
    v2f a0, a1;
    a0[0] = hrow[k0];
    a0[1] = hrow[k0 + 1];
    a1[0] = hrow[k1];
    a1[1] = half ? 0.0f : hrow[5];      // K=7 is zero padding

    float* myl = &lds[wave][0];

    #pragma unroll
    for (int nt = 0; nt < 4; ++nt) {
        const int ncol = nt * 16 + lm;
        // B fragments (4x16, K across half-wave like A, N across lanes)
        v2f bb0, bb1;
        bb0[0] = W1[(k0 + 0) * HID + ncol];
        bb0[1] = W1[(k0 + 1) * HID + ncol];
        bb1[0] = W1[k1 * HID + ncol];
        bb1[1] = half ? 0.0f : W1[5 * HID + ncol];

        v8f acc = {};
        acc = __builtin_amdgcn_wmma_f32_16x16x4_f32(false, a0, false, bb0,
                                                    (short)0, acc, false, false);
        acc = __builtin_amdgcn_wmma_f32_16x16x4_f32(false, a1, false, bb1,
                                                    (short)0, acc, false, false);
        // C/D layout: vgpr v -> row (v + 8*half), lane lm -> col ncol
        const float bias = b1v[ncol];
        #pragma unroll
        for (int v = 0; v < 8; ++v) {
            float pre = acc[v] + bias;
            // SiLU with hardware rcp (1-ulp): avoids the 9-instr IEEE div chain
            float sv  = pre * __builtin_amdgcn_rcpf(1.0f + __expf(-pre));
            myl[(v + 8 * half) * 68 + ncol] = sv;
        }
    }
    __syncthreads();

    // Layer 2: lane L -> (row m = L&15, out col p = L>>4); 64-MAC dot
    const int m = lane & 15;
    const int p = lane >> 4;
    float sum = 0.0f;
    #pragma unroll
    for (int n = 0; n < HID; ++n)
        sum += myl[m * 68 + n] * W2[n * 2 + p];
    sum += b2v[p];
    rec[(size_t)(row0 + m) * 8 + 3 + p] = sum;
}

// ---------------------------------------------------------------------------
// Kernel C: streaming edge-feature kernel (bandwidth bound, 128 MB store).
// One block per (b,i); thread j computes edge (b,i,j); two coalesced,
// non-temporal b128 stores per thread. Hardware sqrt/rcp keeps VALU well
// under the 23.3 TB/s store floor.
// ---------------------------------------------------------------------------
__global__ void edge_kernel(const float* __restrict__ rec,
                            const float* __restrict__ pose,
                            float* __restrict__ out) {
    const int bi = blockIdx.x;          // b*512 + i
    const int b  = bi >> 9;
    const int j  = threadIdx.x;         // 0..511

    // Row-node (i) data: uniform per block
    const float* ri = rec + (size_t)bi * 8;
    const float xi0 = ri[0], xi1 = ri[1], xi2 = ri[2];
    const float hi0 = ri[3], hi1 = ri[4];
    const float* pp = pose + (size_t)bi * 9;   // pose_b[b,i,m,k], row-major m,k
    const float p00 = pp[0], p01 = pp[1], p02 = pp[2];
    const float p10 = pp[3], p11 = pp[4], p12 = pp[5];
    const float p20 = pp[6], p21 = pp[7], p22 = pp[8];

    // Column-node (j) data: packed 32B record -> b128 + b32 (stays L2/L0 hot)
    const float* rjb = rec + (size_t)(b * NN) * 8;
    const v4f   lo  = *(const v4f*)(rjb + (size_t)j * 8);    // {xc0,xc1,xc2,he0}
    const float hj1 = rjb[(size_t)j * 8 + 4];

    const float d0 = xi0 - lo.x;
    const float d1 = xi1 - lo.y;
    const float d2 = xi2 - lo.z;
    const float radial = d0 * d0 + d1 * d1 + d2 * d2;
    // 1 / (sqrt(radial+eps) + 1): single v_sqrt_f32 + single v_rcp_f32
    const float s   = __builtin_amdgcn_sqrtf(radial + EPSF);
    const float inv = __builtin_amdgcn_rcpf(s + 1.0f);       // NORM_CONSTANT = 1
    const float c0 = d0 * inv, c1 = d1 * inv, c2 = d2 * inv;
    // tc[m] = sum_k cd[k] * pose[m][k]
    const float t0 = c0 * p00 + c1 * p01 + c2 * p02;
    const float t1 = c0 * p10 + c1 * p11 + c2 * p12;
    const float t2 = c0 * p20 + c1 * p21 + c2 * p22;

    v4f* o = (v4f*)(out + ((size_t)bi * NN + j) * 8);
    v4f o0; o0.x = radial; o0.y = t0;  o0.z = t1;   o0.w = t2;
    v4f o1; o1.x = hi0;    o1.y = hi1; o1.z = lo.w; o1.w = hj1;
    __builtin_nontemporal_store(o0, o);
    __builtin_nontemporal_store(o1, o + 1);
}

// ---------------------------------------------------------------------------
extern "C" void kernel_launch(void* const* d_in, const int* in_sizes, int n_in,
                              void* d_out, int out_size, void* d_ws, size_t ws_size,
                              hipStream_t stream) {
    (void)in_sizes; (void)n_in; (void)out_size; (void)ws_size;
    const float* x    = (const float*)d_in[0];
    const float* h    = (const float*)d_in[1];
    const float* pose = (const float*)d_in[2];
    const float* mask = (const float*)d_in[3];
    const float* W1   = (const float*)d_in[4];
    const float* b1   = (const float*)d_in[5];
    const float* W2   = (const float*)d_in[6];
    const float* b2   = (const float*)d_in[7];
    float* out = (float*)d_out;
    float* rec = (float*)d_ws;          // 8192 nodes * 8 floats = 256 KB

    center_kernel<<<BS, NN, 0, stream>>>(x, mask, rec);
    mlp_kernel<<<(BS * NN) / (16 * 8), 256, 0, stream>>>(h, W1, b1, W2, b2, rec);
    edge_kernel<<<BS * NN, NN, 0, stream>>>(rec, pose, out);
}